// CSLoss_41223096107308
// MI455X (gfx1250) — compile-verified
//
#include <hip/hip_runtime.h>
#include <hip/hip_bf16.h>
#include <math.h>

typedef float v2f __attribute__((ext_vector_type(2)));
typedef float v8f __attribute__((ext_vector_type(8)));

#define EMB 128
#define NSAMP 64
#define NROWS 16384
#define TILES (NROWS / 16)   // 1024 tiles of 16 rows
#define WAVES_PER_BLOCK 4

// log_sigmoid(x) = min(x,0) - log1p(exp(-|x|))
__device__ __forceinline__ float log_sigmoid(float x) {
    return fminf(x, 0.0f) - __logf(1.0f + __expf(-fabsf(x)));
}

__global__ __launch_bounds__(128) void csloss_wmma_kernel(
    const float* __restrict__ embs,     // [NROWS, EMB]
    const float* __restrict__ weights,  // [NUM_NODES, EMB]
    const int*   __restrict__ label,    // [NROWS]
    const int*   __restrict__ negs,     // [NROWS, NSAMP]
    float*       __restrict__ partials) // [TILES]
{
    const int lane = threadIdx.x & 31;
    const int wave = threadIdx.x >> 5;
    const int tile = blockIdx.x;
    const int i0   = tile * 16;

    const int mrow = lane & 15;        // tile row / B column owned by this lane
    const int half = lane >> 4;        // K-half select per WMMA fragment layout
    const int row  = i0 + mrow;

    // ---- Load A tile (embs rows i0..i0+15, K=0..127) into registers.
    // f32 WMMA 16x16x4 A layout: lanes 0-15 hold K={0,1}, lanes 16-31 K={2,3}.
    const float* arow = embs + (size_t)row * EMB;
    v2f A[32];
#pragma unroll
    for (int t = 0; t < 32; ++t)
        A[t] = *(const v2f*)(arow + 4 * t + 2 * half);

    // Diagonal extraction map for the 16x16 f32 accumulator:
    // lanes 0-7 own D[lane] (M=N=lane); lanes 24-31 own D[lane-24] (M=N=lane-16).
    const int  sel    = (lane < 8) ? lane : (lane - 24);
    const bool active = (lane < 8) || (lane >= 24);

    float partial = 0.0f;

    // Chains: c in {wave, wave+4, ...}; c<64 -> negative slot c, c==64 -> positive.
    for (int c = wave; c < NSAMP + 1; c += WAVES_PER_BLOCK) {
        const int widx = (c < NSAMP) ? negs[(size_t)row * NSAMP + c] : label[row];
        const float* wrow = weights + (size_t)widx * EMB;

        // Prefetch the next chain's gathered row (512B, 16B stride per lane).
        const int cn = c + WAVES_PER_BLOCK;
        if (cn <= NSAMP) {
            const int nidx = (cn < NSAMP) ? negs[(size_t)row * NSAMP + cn] : label[row];
            const float* nrow = weights + (size_t)nidx * EMB;
            __builtin_prefetch(nrow + lane * 4, 0, 0);
        }

        // ---- Load B fragments: column mrow = gathered weight row.
        v2f B[32];
#pragma unroll
        for (int t = 0; t < 32; ++t)
            B[t] = *(const v2f*)(wrow + 4 * t + 2 * half);

        // ---- 32-step f32 WMMA chain over K=128.
        v8f D = {};
#pragma unroll
        for (int t = 0; t < 32; ++t)
            D = __builtin_amdgcn_wmma_f32_16x16x4_f32(
                    /*neg_a=*/false, A[t], /*neg_b=*/false, B[t],
                    /*c_mod=*/(short)0, D, /*reuse_a=*/false, /*reuse_b=*/false);

        // ---- Diagonal pick + log-sigmoid accumulate (negatives: flip sign).
        float x = D[0];
#pragma unroll
        for (int v = 1; v < 8; ++v)
            x = (sel == v) ? D[v] : x;

        const float score = (c < NSAMP) ? -x : x;
        const float ls = log_sigmoid(score);
        if (active) partial += ls;
    }

    // ---- Wave reduce (32 lanes), then deterministic per-block combine.
#pragma unroll
    for (int o = 16; o > 0; o >>= 1)
        partial += __shfl_xor(partial, o, 32);

    __shared__ float wsum[WAVES_PER_BLOCK];
    if (lane == 0) wsum[wave] = partial;
    __syncthreads();
    if (threadIdx.x == 0) {
        float s = 0.0f;
#pragma unroll
        for (int w = 0; w < WAVES_PER_BLOCK; ++w) s += wsum[w];
        partials[tile] = s;
    }
}

__global__ __launch_bounds__(256) void csloss_reduce_kernel(
    const float* __restrict__ partials, float* __restrict__ out)
{
    __shared__ float sm[256];
    float s = 0.0f;
    for (int i = threadIdx.x; i < TILES; i += 256) s += partials[i];
    sm[threadIdx.x] = s;
    __syncthreads();
    for (int o = 128; o > 0; o >>= 1) {
        if ((int)threadIdx.x < o) sm[threadIdx.x] += sm[threadIdx.x + o];
        __syncthreads();
    }
    if (threadIdx.x == 0) out[0] = sm[0] * (-1.0f / (float)NROWS);
}

extern "C" void kernel_launch(void* const* d_in, const int* in_sizes, int n_in,
                              void* d_out, int out_size, void* d_ws, size_t ws_size,
                              hipStream_t stream) {
    // d_in order: input (unused), embs, weights, label, negs
    const float* embs    = (const float*)d_in[1];
    const float* weights = (const float*)d_in[2];
    const int*   label   = (const int*)d_in[3];
    const int*   negs    = (const int*)d_in[4];
    float* out      = (float*)d_out;
    float* partials = (float*)d_ws;   // TILES floats = 4 KB

    csloss_wmma_kernel<<<dim3(TILES), dim3(32 * WAVES_PER_BLOCK), 0, stream>>>(
        embs, weights, label, negs, partials);
    csloss_reduce_kernel<<<dim3(1), dim3(256), 0, stream>>>(partials, out);
}